// SelfLoss_24953759989822
// MI455X (gfx1250) — compile-verified
//
#include <hip/hip_runtime.h>
#include <math.h>

typedef __attribute__((ext_vector_type(2))) float v2f;
typedef __attribute__((ext_vector_type(8))) float v8f;

#define IMG_H 120
#define IMG_W 160
#define NPTS  (IMG_H * IMG_W)     /* 19200 */
#define TILE  16
#define NTILES (NPTS / TILE)      /* 1200 */

#define CAM_FX 572.4114f
#define CAM_FY 573.57043f
#define CAM_CX 325.2611f
#define CAM_CY 242.04899f

// Workspace layout in floats:
//  [0,      4N)  src_pts  (x,y,z,0)
//  [4N,     8N)  tgt_pts  (x,y,z,0)
//  [8N,     9N)  src_norm2
//  [9N,    10N)  tgt_norm2
//  [10N,   11N)  dmin_src
//  [11N,   12N)  dmin_tgt

__global__ void backproject_kernel(const float* __restrict__ depth,
                                   float* __restrict__ pts,
                                   float* __restrict__ norm2) {
    int i = blockIdx.x * blockDim.x + threadIdx.x;
    if (i >= NPTS) return;
    int r = i / IMG_W;
    int c = i - r * IMG_W;
    float d = depth[i];
    float x = ((float)c - CAM_CX) * d / CAM_FX;
    float y = ((float)r - CAM_CY) * d / CAM_FY;
    pts[i * 4 + 0] = x;
    pts[i * 4 + 1] = y;
    pts[i * 4 + 2] = d;
    pts[i * 4 + 3] = 0.0f;   // K=4 zero pad so WMMA dot == 3D dot
    norm2[i] = x * x + y * y + d * d;
}

// One wave32 owns 16 rows of A; sweeps all 1200 16-col tiles of B with
// V_WMMA_F32_16X16X4_F32. Running per-lane min of (|b|^2 - 2 a.b); |a|^2
// added after the cross-lane reduction; clamp to >=0 at the end.
__global__ void chamfer_min_kernel(const float* __restrict__ a_pts,
                                   const float* __restrict__ b_pts,
                                   const float* __restrict__ a_norm2,
                                   const float* __restrict__ b_norm2,
                                   float* __restrict__ dmin) {
    int wave = (blockIdx.x * blockDim.x + threadIdx.x) >> 5;
    int lane = threadIdx.x & 31;
    if (wave >= NTILES) return;          // whole-wave exit only (grid sized exactly)

    int rowbase = wave * TILE;
    int m       = lane & 15;             // A-row / B-col index within tile
    int khalf   = (lane >> 4) << 1;      // lanes 0-15 -> K={0,1}, lanes 16-31 -> K={2,3}

    // A fragment (16x4, 32-bit): row = rowbase+m, two consecutive K components
    v2f a = *(const v2f*)(a_pts + (size_t)(rowbase + m) * 4 + khalf);

    v8f rmin;
#pragma unroll
    for (int i = 0; i < 8; ++i) rmin[i] = 3.0e38f;

    for (int t = 0; t < NTILES; ++t) {
        int col = t * TILE + m;
        // B fragment (4x16, 32-bit): col = N, same K-pair split as A
        v2f  b  = *(const v2f*)(b_pts + (size_t)col * 4 + khalf);
        float bn = b_norm2[col];

        v8f c = {};
        v8f dot = __builtin_amdgcn_wmma_f32_16x16x4_f32(
            /*neg_a=*/false, a, /*neg_b=*/false, b,
            /*c_mod=*/(short)0, c, /*reuse_a=*/false, /*reuse_b=*/false);

#pragma unroll
        for (int i = 0; i < 8; ++i) {
            float v = fmaf(-2.0f, dot[i], bn);   // |b|^2 - 2 a.b
            rmin[i] = fminf(rmin[i], v);
        }
    }

    // Min over the 16 lanes of each half (same M set per half on wave32)
#pragma unroll
    for (int off = 1; off < 16; off <<= 1) {
#pragma unroll
        for (int i = 0; i < 8; ++i) {
            float o = __shfl_xor(rmin[i], off, 32);
            rmin[i] = fminf(rmin[i], o);
        }
    }

    // C/D layout: VGPR i -> M=i (lanes 0-15), M=8+i (lanes 16-31)
    if (m == 0) {                         // lanes 0 and 16
        int mofs = (lane >> 4) * 8;
#pragma unroll
        for (int i = 0; i < 8; ++i) {
            int row = rowbase + mofs + i;
            float d = a_norm2[row] + rmin[i];
            dmin[row] = fmaxf(d, 0.0f);
        }
    }
}

// Single-block finalize: chamfer means + constant-weight BCE mask loss.
__global__ void finalize_kernel(const float* __restrict__ dmin_src,
                                const float* __restrict__ dmin_tgt,
                                const float* __restrict__ pred_PM,
                                const float* __restrict__ pred_Ms,
                                float* __restrict__ out) {
    __shared__ float red[5][256];
    int t = threadIdx.x;

    float cham = 0.0f, pos_sum = 0.0f, neg_sum = 0.0f;
    float pos_cnt = 0.0f, neg_cnt = 0.0f;
    const float wgt = 1.3989422804014327f;   // 1/sqrt(2*pi) + 1 (mask_edge==0 bug)

    for (int i = t; i < NPTS; i += 256) {
        cham += dmin_src[i] + dmin_tgt[i];

        float p = pred_PM[i];
        p = fminf(fmaxf(p, 1e-7f), 1.0f - 1e-7f);
        float m = pred_Ms[i];
        if (m > 0.0f) {
            pos_sum += -m * logf(p) * wgt;
            pos_cnt += 1.0f;
        } else {
            neg_sum += -logf(1.0f - p) * wgt;
            neg_cnt += 1.0f;
        }
    }

    red[0][t] = cham;
    red[1][t] = pos_sum;
    red[2][t] = neg_sum;
    red[3][t] = pos_cnt;
    red[4][t] = neg_cnt;

    for (int s = 128; s > 0; s >>= 1) {
        __syncthreads();
        if (t < s) {
#pragma unroll
            for (int j = 0; j < 5; ++j) red[j][t] += red[j][t + s];
        }
    }
    __syncthreads();

    if (t == 0) {
        float inv_n = 1.0f / (float)NPTS;
        out[0] = red[0][0] * inv_n;          // mean(src mins) + mean(tgt mins)
        float np = red[3][0], nn = red[4][0];
        float loss = (np > 0.0f) ? red[1][0] / fmaxf(np, 1.0f) : 0.0f;
        loss      += (nn > 0.0f) ? red[2][0] / fmaxf(nn, 1.0f) : 0.0f;
        out[1] = loss;
    }
}

extern "C" void kernel_launch(void* const* d_in, const int* in_sizes, int n_in,
                              void* d_out, int out_size, void* d_ws, size_t ws_size,
                              hipStream_t stream) {
    const float* pred_PM   = (const float*)d_in[0];
    const float* pred_Ms   = (const float*)d_in[1];
    const float* depth_src = (const float*)d_in[2];
    const float* depth_tgt = (const float*)d_in[3];
    float* out = (float*)d_out;

    float* ws = (float*)d_ws;
    float* src_pts   = ws;
    float* tgt_pts   = ws + 4 * NPTS;
    float* src_norm2 = ws + 8 * NPTS;
    float* tgt_norm2 = ws + 9 * NPTS;
    float* dmin_src  = ws + 10 * NPTS;
    float* dmin_tgt  = ws + 11 * NPTS;

    (void)in_sizes; (void)n_in; (void)out_size; (void)ws_size;

    int bp_blocks = (NPTS + 255) / 256;
    backproject_kernel<<<bp_blocks, 256, 0, stream>>>(depth_src, src_pts, src_norm2);
    backproject_kernel<<<bp_blocks, 256, 0, stream>>>(depth_tgt, tgt_pts, tgt_norm2);

    // 1200 waves -> 150 blocks of 256 threads (8 waves each), no partial waves.
    chamfer_min_kernel<<<NTILES / 8, 256, 0, stream>>>(src_pts, tgt_pts, src_norm2, tgt_norm2, dmin_src);
    chamfer_min_kernel<<<NTILES / 8, 256, 0, stream>>>(tgt_pts, src_pts, tgt_norm2, src_norm2, dmin_tgt);

    finalize_kernel<<<1, 256, 0, stream>>>(dmin_src, dmin_tgt, pred_PM, pred_Ms, out);
}